// ProteinGNNEncoder_29008209117390
// MI455X (gfx1250) — compile-verified
//
#include <hip/hip_runtime.h>
#include <hip/hip_bf16.h>

typedef __attribute__((ext_vector_type(2))) float v2f;
typedef __attribute__((ext_vector_type(8))) float v8f;

#define HDIM 128  // hidden width (H in reference)

// ---------------------------------------------------------------------------
// Degree / normalization:  deg[n] = 1 (self loop) + #edges with dst==n
// dinv[n] = deg^-1/2
// ---------------------------------------------------------------------------
__global__ void init_deg_kernel(float* __restrict__ deg, int n) {
    int i = blockIdx.x * blockDim.x + threadIdx.x;
    if (i < n) deg[i] = 1.0f;
}

__global__ void count_deg_kernel(const int* __restrict__ dst,
                                 float* __restrict__ deg, int E) {
    int i = blockIdx.x * blockDim.x + threadIdx.x;
    if (i < E) unsafeAtomicAdd(&deg[dst[i]], 1.0f);
}

__global__ void finish_deg_kernel(float* __restrict__ dinv, int n) {
    int i = blockIdx.x * blockDim.x + threadIdx.x;
    if (i < n) dinv[i] = rsqrtf(dinv[i]);
}

// ---------------------------------------------------------------------------
// FP32 WMMA GEMM, fused with per-row dinv scaling:
//   out[n, :] = (X[n, :] @ W) * dinv[n]          X: nrows x K, W: K x 128
// One wave computes one 16x16 tile using V_WMMA_F32_16X16X4_F32.
// Fragment layouts (CDNA5 ISA 7.12.2, wave32):
//   A 16x4 : lane l (l<16): vgpr v = A[M=l][K=v];   l>=16: A[M=l-16][K=v+2]
//   B 4x16 : lane l (l<16): vgpr v = B[K=v][N=l];   l>=16: B[K=v+2][N=l-16]
//   C/D    : lane l, vgpr v = D[M = v + 8*(l/16)][N = l%16]
// ---------------------------------------------------------------------------
template <int K>
__global__ void gemm_scale_kernel(const float* __restrict__ X,
                                  const float* __restrict__ W,
                                  const float* __restrict__ dinv,
                                  float* __restrict__ out, int nrows) {
    const int colTiles = HDIM / 16;  // 8
    int wave = (blockIdx.x * blockDim.x + threadIdx.x) >> 5;
    int lane = threadIdx.x & 31;
    int rowTile = wave / colTiles;
    int colTile = wave % colTiles;
    int row0 = rowTile * 16;
    if (row0 >= nrows) return;  // wave-uniform: EXEC all-ones inside
    int col0 = colTile * 16;

    int lm = lane & 15;  // 0..15
    int lh = lane >> 4;  // 0 or 1

    v8f acc = {};
    const float* arow = X + (size_t)(row0 + lm) * K + 2 * lh;
    const float* bcol = W + (size_t)(2 * lh) * HDIM + col0 + lm;
    for (int kc = 0; kc < K; kc += 4) {
        // A fragment: X[row0+lm][kc + 2*lh + {0,1}]  (contiguous pair)
        v2f a;
        a.x = arow[kc + 0];
        a.y = arow[kc + 1];
        // B fragment: W[kc + 2*lh + {0,1}][col0+lm]  (stride HDIM)
        v2f b;
        b.x = bcol[(size_t)kc * HDIM];
        b.y = bcol[(size_t)(kc + 1) * HDIM];
        acc = __builtin_amdgcn_wmma_f32_16x16x4_f32(
            /*neg_a=*/false, a, /*neg_b=*/false, b,
            /*c_mod=*/(short)0, acc, /*reuse_a=*/false, /*reuse_b=*/false);
    }

#pragma unroll
    for (int v = 0; v < 8; ++v) {
        int r = row0 + v + 8 * lh;
        out[(size_t)r * HDIM + col0 + lm] = acc[v] * dinv[r];
    }
}

// ---------------------------------------------------------------------------
// Edge scatter: acc[dst] += hs[src]   (hs already scaled by dinv[src]).
// 128 consecutive threads handle one edge -> contiguous 512B row gather and
// contiguous atomic-add row, lowering to global_atomic_add_f32.
// ---------------------------------------------------------------------------
__global__ void scatter_add_kernel(const float* __restrict__ hs,
                                   const int* __restrict__ src,
                                   const int* __restrict__ dst,
                                   float* __restrict__ acc, int E) {
    size_t t = (size_t)blockIdx.x * blockDim.x + threadIdx.x;
    int e = (int)(t >> 7);
    int f = (int)(t & (HDIM - 1));
    if (e >= E) return;
    int s = src[e];
    int d = dst[e];
    unsafeAtomicAdd(&acc[(size_t)d * HDIM + f], hs[(size_t)s * HDIM + f]);
}

// out = relu(acc * dinv[node] + bias)   (in-place safe)
__global__ void finish_layer_kernel(const float* __restrict__ acc,
                                    const float* __restrict__ dinv,
                                    const float* __restrict__ bias,
                                    float* __restrict__ out, int n) {
    size_t t = (size_t)blockIdx.x * blockDim.x + threadIdx.x;
    if (t >= (size_t)n * HDIM) return;
    int node = (int)(t >> 7);
    int f = (int)(t & (HDIM - 1));
    out[t] = fmaxf(acc[t] * dinv[node] + bias[f], 0.0f);
}

// ---------------------------------------------------------------------------
// Global mean pool
// ---------------------------------------------------------------------------
__global__ void pool_zero_kernel(float* __restrict__ pooled,
                                 float* __restrict__ cnts, int B) {
    int i = blockIdx.x * blockDim.x + threadIdx.x;
    if (i < B * HDIM) pooled[i] = 0.0f;
    if (i < B) cnts[i] = 0.0f;
}

__global__ void pool_feat_kernel(const float* __restrict__ x,
                                 const int* __restrict__ batch,
                                 float* __restrict__ pooled, int n) {
    size_t t = (size_t)blockIdx.x * blockDim.x + threadIdx.x;
    if (t >= (size_t)n * HDIM) return;
    int node = (int)(t >> 7);
    int f = (int)(t & (HDIM - 1));
    unsafeAtomicAdd(&pooled[(size_t)batch[node] * HDIM + f], x[t]);
}

__global__ void pool_cnt_kernel(const int* __restrict__ batch,
                                float* __restrict__ cnts, int n) {
    int i = blockIdx.x * blockDim.x + threadIdx.x;
    if (i < n) unsafeAtomicAdd(&cnts[batch[i]], 1.0f);
}

__global__ void pool_div_kernel(float* __restrict__ pooled,
                                const float* __restrict__ cnts, int B) {
    int i = blockIdx.x * blockDim.x + threadIdx.x;
    if (i < B * HDIM) pooled[i] /= fmaxf(cnts[i >> 7], 1.0f);
}

// ---------------------------------------------------------------------------
// MLP head (tiny: 64x128 @ 128x128 and 64x128 @ 128x64) — plain VALU FMA
// ---------------------------------------------------------------------------
__global__ void head1_kernel(const float* __restrict__ pooled,
                             const float* __restrict__ W3,
                             const float* __restrict__ b3,
                             float* __restrict__ t1, int B) {
    int t = blockIdx.x * blockDim.x + threadIdx.x;
    if (t >= B * HDIM) return;
    int b = t >> 7;
    int j = t & (HDIM - 1);
    float s = b3[j];
    const float* p = pooled + (size_t)b * HDIM;
#pragma unroll 4
    for (int k = 0; k < HDIM; ++k) s = fmaf(p[k], W3[(size_t)k * HDIM + j], s);
    t1[t] = fmaxf(s, 0.0f);
}

__global__ void head2_kernel(const float* __restrict__ t1,
                             const float* __restrict__ W4,
                             const float* __restrict__ b4,
                             float* __restrict__ out, int B) {
    int t = blockIdx.x * blockDim.x + threadIdx.x;
    if (t >= B * 64) return;
    int b = t >> 6;
    int j = t & 63;
    float s = b4[j];
    const float* p = t1 + (size_t)b * HDIM;
#pragma unroll 4
    for (int k = 0; k < HDIM; ++k) s = fmaf(p[k], W4[(size_t)k * 64 + j], s);
    out[t] = s;
}

// ---------------------------------------------------------------------------
extern "C" void kernel_launch(void* const* d_in, const int* in_sizes, int n_in,
                              void* d_out, int out_size, void* d_ws, size_t ws_size,
                              hipStream_t stream) {
    const float* x   = (const float*)d_in[0];
    const int*   ei  = (const int*)d_in[1];   // [2, E]: row0=src, row1=dst
    const int*   bat = (const int*)d_in[2];
    const float* W1  = (const float*)d_in[3];
    const float* b1  = (const float*)d_in[4];
    const float* W2  = (const float*)d_in[5];
    const float* b2  = (const float*)d_in[6];
    const float* W3  = (const float*)d_in[7];
    const float* b3  = (const float*)d_in[8];
    const float* W4  = (const float*)d_in[9];
    const float* b4  = (const float*)d_in[10];

    const int IN = 64;
    const int N  = in_sizes[0] / IN;
    const int E  = in_sizes[1] / 2;
    const int B  = out_size / 64;
    const int* src = ei;
    const int* dst = ei + E;

    // workspace layout (floats)
    float* ws     = (float*)d_ws;
    size_t NH     = (size_t)N * HDIM;
    float* dinv   = ws;
    float* buf0   = ws + (((size_t)N + 255) & ~(size_t)255);
    float* buf1   = buf0 + NH;
    float* pooled = buf1 + NH;
    float* cnts   = pooled + (size_t)B * HDIM;
    float* t1     = cnts + B;
    float* out    = (float*)d_out;

    const int T = 256;
    auto blocks = [](size_t work, int t) { return (unsigned)((work + t - 1) / t); };

    // --- gcn_norm ---
    init_deg_kernel<<<blocks(N, T), T, 0, stream>>>(dinv, N);
    count_deg_kernel<<<blocks(E, T), T, 0, stream>>>(dst, dinv, E);
    finish_deg_kernel<<<blocks(N, T), T, 0, stream>>>(dinv, N);

    const int rowTiles = (N + 15) / 16;
    const size_t gemmThreads = (size_t)rowTiles * (HDIM / 16) * 32;

    // --- layer 1: hs = (x @ W1) * dinv ---
    gemm_scale_kernel<64><<<blocks(gemmThreads, T), T, 0, stream>>>(x, W1, dinv, buf0, N);
    // self-loop term: acc = hs
    hipMemcpyAsync(buf1, buf0, NH * sizeof(float), hipMemcpyDeviceToDevice, stream);
    scatter_add_kernel<<<blocks((size_t)E * HDIM, T), T, 0, stream>>>(buf0, src, dst, buf1, E);
    finish_layer_kernel<<<blocks(NH, T), T, 0, stream>>>(buf1, dinv, b1, buf1, N);

    // --- layer 2: hs2 = (x1 @ W2) * dinv ---
    gemm_scale_kernel<128><<<blocks(gemmThreads, T), T, 0, stream>>>(buf1, W2, dinv, buf0, N);
    hipMemcpyAsync(buf1, buf0, NH * sizeof(float), hipMemcpyDeviceToDevice, stream);
    scatter_add_kernel<<<blocks((size_t)E * HDIM, T), T, 0, stream>>>(buf0, src, dst, buf1, E);
    finish_layer_kernel<<<blocks(NH, T), T, 0, stream>>>(buf1, dinv, b2, buf1, N);

    // --- global mean pool ---
    pool_zero_kernel<<<blocks((size_t)B * HDIM, T), T, 0, stream>>>(pooled, cnts, B);
    pool_feat_kernel<<<blocks(NH, T), T, 0, stream>>>(buf1, bat, pooled, N);
    pool_cnt_kernel<<<blocks(N, T), T, 0, stream>>>(bat, cnts, N);
    pool_div_kernel<<<blocks((size_t)B * HDIM, T), T, 0, stream>>>(pooled, cnts, B);

    // --- MLP head ---
    head1_kernel<<<blocks((size_t)B * HDIM, T), T, 0, stream>>>(pooled, W3, b3, t1, B);
    head2_kernel<<<blocks((size_t)B * 64, T), T, 0, stream>>>(t1, W4, b4, out, B);
}